// BaselineGCN_16149077033546
// MI455X (gfx1250) — compile-verified
//
#include <hip/hip_runtime.h>

typedef __attribute__((ext_vector_type(2))) float v2f;
typedef __attribute__((ext_vector_type(4))) float v4f;
typedef __attribute__((ext_vector_type(8))) float v8f;

// ---------------------------------------------------------------- utilities
__global__ void fill_kernel(float* __restrict__ p, float v, long long n) {
  long long i = (long long)blockIdx.x * blockDim.x + threadIdx.x;
  long long stride = (long long)gridDim.x * blockDim.x;
  for (; i < n; i += stride) p[i] = v;
}

__global__ void degree_kernel(const long long* __restrict__ dst,
                              float* __restrict__ deg, int E) {
  int e = blockIdx.x * blockDim.x + threadIdx.x;
  if (e < E) atomicAdd(&deg[(int)dst[e]], 1.0f);
}

__global__ void rsqrt_kernel(float* __restrict__ deg, int n) {
  int i = blockIdx.x * blockDim.x + threadIdx.x;
  if (i < n) deg[i] = rsqrtf(deg[i]);   // deg >= 1 always (self-loop)
}

// enorm[e] = dinv[src[e]] * dinv[dst[e]]  (one-time random gathers)
__global__ void edgenorm_kernel(const long long* __restrict__ src,
                                const long long* __restrict__ dst,
                                const float* __restrict__ dinv,
                                float* __restrict__ enorm, int E) {
  int e = blockIdx.x * blockDim.x + threadIdx.x;
  if (e < E) enorm[e] = dinv[(int)src[e]] * dinv[(int)dst[e]];
}

// ---------------------------------------------------------------- WMMA GEMM
// C[M,Nout] = A[M,K] * B[K,Nout], all row-major fp32.
// One wave computes a 16x16 tile of C via V_WMMA_F32_16X16X4_F32.
// blockDim.x = 32 * (Nout/16): wave w handles column tile w*16.
// A layout (16x4): lanes 0-15 rows M=0..15 with VGPR0/1 = K0/K1;
//                  lanes 16-31 same rows with K2/K3. B mirrors on N.
__global__ void gemm_wmma_f32(const float* __restrict__ A,
                              const float* __restrict__ B,
                              float* __restrict__ C,
                              int M, int K, int Nout) {
  int lane = threadIdx.x & 31;
  int wave = threadIdx.x >> 5;
  int m0   = blockIdx.x * 16;
  int n0   = wave * 16;
  int half = lane >> 4;      // 0: K pair (0,1), 1: K pair (2,3)
  int l16  = lane & 15;

  int row = m0 + l16;
  if (row >= M) row = M - 1;                       // keep EXEC full on edge tiles
  const float* arow = A + (size_t)row * K + half * 2;
  const float* bptr = B + (size_t)(half * 2) * Nout + n0 + l16;

  v8f acc = {};
  for (int k = 0; k < K; k += 4) {
    v2f a; a.x = arow[k];                 a.y = arow[k + 1];
    v2f b; b.x = bptr[(size_t)k * Nout];  b.y = bptr[(size_t)(k + 1) * Nout];
    acc = __builtin_amdgcn_wmma_f32_16x16x4_f32(
        /*neg_a=*/false, a, /*neg_b=*/false, b,
        /*c_mod=*/(short)0, acc, /*reuse_a=*/false, /*reuse_b=*/false);
  }

  // C/D layout: VGPR r -> row m0 + r (lanes 0-15) or m0 + r + 8 (lanes 16-31),
  // column n0 + (lane & 15).
  int crow0 = m0 + half * 8;
  float* cp = C + (size_t)crow0 * Nout + n0 + l16;
#pragma unroll
  for (int r = 0; r < 8; ++r)
    if (crow0 + r < M) cp[(size_t)r * Nout] = acc[r];
}

// ---------------------------------------------------------------- scatter
// One thread handles (edge e, feature quad q): 2^lq quads per edge, F = 2^(lq+2).
// float4 gather of h[src] + 4 f32 atomics into acc[dst]; consecutive threads
// share an edge -> coalesced b128 loads, clustered atomics resolving in the
// 192MB L2 (h and acc are fully L2-resident).
__global__ void scatter_kernel(const long long* __restrict__ src,
                               const long long* __restrict__ dst,
                               const float* __restrict__ enorm,
                               const float* __restrict__ h,
                               float* __restrict__ acc,
                               long long total, int lq, int F) {
  long long tid = (long long)blockIdx.x * blockDim.x + threadIdx.x;
  if (tid >= total) return;
  long long e = tid >> lq;
  int q = ((int)(tid & ((1LL << lq) - 1))) << 2;
  int s = (int)src[e];
  int d = (int)dst[e];
  float norm = enorm[e];
  v4f hv = *(const v4f*)(h + (size_t)s * F + q);
  float* ap = acc + (size_t)d * F + q;
  atomicAdd(ap + 0, hv.x * norm);
  atomicAdd(ap + 1, hv.y * norm);
  atomicAdd(ap + 2, hv.z * norm);
  atomicAdd(ap + 3, hv.w * norm);
}

// acc[n,f] = maybe_relu(acc[n,f] + h[n,f]*dinv[n]^2 + bias[f])   (F = 2^lf)
__global__ void finalize_kernel(float* __restrict__ acc,
                                const float* __restrict__ h,
                                const float* __restrict__ dinv,
                                const float* __restrict__ bias,
                                long long total, int lf, int relu) {
  long long i = (long long)blockIdx.x * blockDim.x + threadIdx.x;
  if (i >= total) return;
  long long n = i >> lf;
  int f = (int)(i & ((1LL << lf) - 1));
  float di = dinv[n];
  float v = acc[i] + h[i] * di * di + bias[f];
  if (relu) v = fmaxf(v, 0.0f);
  acc[i] = v;
}

// ---------------------------------------------------------------- driver
extern "C" void kernel_launch(void* const* d_in, const int* in_sizes, int n_in,
                              void* d_out, int out_size, void* d_ws, size_t ws_size,
                              hipStream_t stream) {
  const float*     x   = (const float*)d_in[0];      // [N,128]
  const long long* ei  = (const long long*)d_in[1];  // [2,E] int64
  const float*     W1  = (const float*)d_in[2];      // [128,64]
  const float*     b1  = (const float*)d_in[3];      // [64]
  const float*     W2  = (const float*)d_in[4];      // [64,32]
  const float*     b2  = (const float*)d_in[5];      // [32]
  float*           out = (float*)d_out;              // [N,32]

  const int NN = in_sizes[0] / 128;   // 100000
  const int E  = in_sizes[1] / 2;     // 1600000
  const long long* src = ei;
  const long long* dst = ei + E;

  // workspace layout: dinv[NN] | enorm[E] | h1[64*NN] | acc1[64*NN]
  // (h2 reuses h1's region after finalize1; 32*NN <= 64*NN)
  float* dinv  = (float*)d_ws;
  float* enorm = dinv + NN;
  float* h1    = enorm + E;
  float* acc1  = h1 + (size_t)64 * NN;
  float* h2    = h1;

  const int B = 256;
  auto blks = [](long long n, int b) { return (unsigned)((n + b - 1) / b); };

  // ---- init: deg = 1 (self-loops), acc1 = 0, out = 0
  fill_kernel<<<blks(NN, B), B, 0, stream>>>(dinv, 1.0f, NN);
  fill_kernel<<<blks((long long)64 * NN, B), B, 0, stream>>>(acc1, 0.0f, (long long)64 * NN);
  fill_kernel<<<blks((long long)out_size, B), B, 0, stream>>>(out, 0.0f, (long long)out_size);

  // ---- degree + symmetric normalization + per-edge norm
  degree_kernel<<<blks(E, B), B, 0, stream>>>(dst, dinv, E);
  rsqrt_kernel<<<blks(NN, B), B, 0, stream>>>(dinv, NN);
  edgenorm_kernel<<<blks(E, B), B, 0, stream>>>(src, dst, dinv, enorm, E);

  // ============================ layer 1: 128 -> 64 ============================
  gemm_wmma_f32<<<blks(NN, 16), 4 * 32, 0, stream>>>(x, W1, h1, NN, 128, 64);

  long long tot1 = (long long)E * 16;                   // 16 quads/edge (F=64)
  scatter_kernel<<<blks(tot1, B), B, 0, stream>>>(src, dst, enorm, h1, acc1, tot1, 4, 64);

  long long ne1 = (long long)NN * 64;
  finalize_kernel<<<blks(ne1, B), B, 0, stream>>>(acc1, h1, dinv, b1, ne1, 6, 1);

  // ============================ layer 2: 64 -> 32 =============================
  gemm_wmma_f32<<<blks(NN, 16), 2 * 32, 0, stream>>>(acc1, W2, h2, NN, 64, 32);

  long long tot2 = (long long)E * 8;                    // 8 quads/edge (F=32)
  scatter_kernel<<<blks(tot2, B), B, 0, stream>>>(src, dst, enorm, h2, out, tot2, 3, 32);

  long long ne2 = (long long)NN * 32;
  finalize_kernel<<<blks(ne2, B), B, 0, stream>>>(out, h2, dinv, b2, ne2, 5, 0);
}